// SDP_10230612099166
// MI455X (gfx1250) — compile-verified
//
#include <hip/hip_runtime.h>

typedef __attribute__((ext_vector_type(2))) float v2f;
typedef __attribute__((ext_vector_type(8))) float v8f;

#define HDIM 128
#define NBATCH 1024
#define WAVES_PER_BLOCK 8
#define PTS_PER_WAVE 16
#define PTS_PER_BLOCK (WAVES_PER_BLOCK * PTS_PER_WAVE) /* 128 */
/* B-fragment LDS layout: sB[k2][j] = float2(W2[2*k2][j], W2[2*k2+1][j]).
   Row stride 288 words (128 float2 = 256 words, +32 pad) so the two lane
   halves (k2 vs k2+1) land on disjoint LDS bank sets for ds_load_b64. */
#define SB_STRIDE 288 /* words per k2 row */

// Branch-free tanh: gfx1250 hardware v_tanh_f32 (TRANS32, co-executes with
// the matrix pipe); fallback is exp-based identity with v_exp_f32.
__device__ __forceinline__ float fast_tanh(float x) {
#if __has_builtin(__builtin_amdgcn_tanhf)
  return __builtin_amdgcn_tanhf(x);
#else
  float e = __expf(2.0f * x);
  return 1.0f - 2.0f / (e + 1.0f);
#endif
}

// ---------------------------------------------------------------------------
// Kernel 1: per-batch counts
// ---------------------------------------------------------------------------
__global__ void sdp_count_kernel(const int* __restrict__ bids,
                                 unsigned int* __restrict__ counts, int n) {
  int i = blockIdx.x * blockDim.x + threadIdx.x;
  if (i < n) atomicAdd(&counts[bids[i]], 1u);
}

// ---------------------------------------------------------------------------
// Kernel 2: fused forward + VJP. One wave = 16 points.
//   forward:  PRE2 = H1 @ W2          (A = h1 frags,  B = W2 tiles)
//   backward: M    = G1 @ W2          (A = g1 frags,  SAME B tiles)
//   s[p]   = sum_j tanh(PRE2+b2)[p,j] * W3[j] + b3
//   jac[p] = sum_j W3[j](1-h2^2)[p,j] * M[p,j]
// ---------------------------------------------------------------------------
__launch_bounds__(256, 1)
__global__ void sdp_main_kernel(const float* __restrict__ features,
                                const int* __restrict__ bids,
                                const float* __restrict__ W1,
                                const float* __restrict__ b1,
                                const float* __restrict__ W2,
                                const float* __restrict__ b2,
                                const float* __restrict__ W3,
                                const float* __restrict__ b3,
                                const unsigned int* __restrict__ counts,
                                float* __restrict__ accum /* [0,B) mean, [B,2B) var */) {
  __shared__ float sB[(HDIM / 2) * SB_STRIDE];  // 72 KB, pre-paired W2 frags
  __shared__ float sW1[HDIM], sb1[HDIM], sb2[HDIM], sW3[HDIM];

  const int tid = threadIdx.x;
  // Stage W2 into fragment-paired layout.
  for (int i = tid; i < HDIM * HDIM; i += 256) {
    const int k = i >> 7, j = i & 127;
    sB[(k >> 1) * SB_STRIDE + 2 * j + (k & 1)] = W2[i];
  }
  if (tid < HDIM) {
    sW1[tid] = W1[tid];
    sb1[tid] = b1[tid];
    sb2[tid] = b2[tid];
    sW3[tid] = W3[tid];
  }
  __syncthreads();

  const int wave = tid >> 5;
  const int lane = tid & 31;
  const int half = lane >> 4;  // 0 or 1
  const int l15  = lane & 15;
  const int base = (blockIdx.x * WAVES_PER_BLOCK + wave) * PTS_PER_WAVE;

  // A-matrix rows = points: lane holds row (lane&15). x0 for that row:
  const float x0 = features[2 * (base + l15) + 0];

  v8f C2[8], CM[8];
#pragma unroll
  for (int t = 0; t < 8; ++t) {
#pragma unroll
    for (int r = 0; r < 8; ++r) { C2[t][r] = 0.0f; CM[t][r] = 0.0f; }
  }

  // K loop over hidden dim of layer-1 output, 4 per WMMA.
  for (int kk = 0; kk < HDIM; kk += 4) {
    // f32 16x4 A layout: lanes 0-15 -> K = kk,kk+1 ; lanes 16-31 -> K = kk+2,kk+3
    const int k0 = kk + 2 * half;
    const int k1 = k0 + 1;
    const float w10 = sW1[k0], w11 = sW1[k1];
    const float h10 = fast_tanh(x0 * w10 + sb1[k0]);
    const float h11 = fast_tanh(x0 * w11 + sb1[k1]);
    v2f Ah; Ah[0] = h10;                      Ah[1] = h11;
    v2f Ag; Ag[0] = (1.0f - h10 * h10) * w10; Ag[1] = (1.0f - h11 * h11) * w11;

    // Pre-paired B fragments: one aligned ds_load_b64 per tile per lane.
    const float* brow = &sB[((kk >> 1) + half) * SB_STRIDE];
#pragma unroll
    for (int t = 0; t < 8; ++t) {
      // f32 4x16 B layout (mirror of A): lane half selects K pair, N = lane&15
      const v2f Bf = *(const v2f*)&brow[2 * (t * 16 + l15)];
      C2[t] = __builtin_amdgcn_wmma_f32_16x16x4_f32(
          false, Ah, false, Bf, (short)0, C2[t], false, false);
      CM[t] = __builtin_amdgcn_wmma_f32_16x16x4_f32(
          false, Ag, false, Bf, (short)0, CM[t], false, false);
    }
  }

  // Epilogue: C/D layout: VGPR r, lanes 0-15 -> point base+r, lanes 16-31 -> base+8+r,
  // column j = t*16 + (lane&15).
  const float b3v = b3[0];
#pragma unroll
  for (int r = 0; r < 8; ++r) {
    float ss = 0.0f, jj = 0.0f;
#pragma unroll
    for (int t = 0; t < 8; ++t) {
      const int j = t * 16 + l15;
      const float h2 = fast_tanh(C2[t][r] + sb2[j]);
      const float w3 = sW3[j];
      ss += h2 * w3;
      jj += (w3 * (1.0f - h2 * h2)) * CM[t][r];
    }
    // reduce across the 16 lanes of each half (j dimension)
#pragma unroll
    for (int m = 1; m < 16; m <<= 1) {
      ss += __shfl_xor(ss, m, 32);
      jj += __shfl_xor(jj, m, 32);
    }
    if (l15 == 0) {
      const int p = base + r + half * 8;
      const int bid = bids[p];
      unsigned int c = counts[bid];
      if (c < 1u) c = 1u;
      const float inv = 1.0f / (float)c;
      const float s = ss + b3v;
      const float stdv = features[2 * p + 1];
      const float jac = jj * inv;
      atomicAdd(&accum[bid], s * inv);
      atomicAdd(&accum[NBATCH + bid], stdv * stdv * jac * jac);
    }
  }
}

// ---------------------------------------------------------------------------
// Kernel 3: interleave [mean, var] into d_out (B,2) row-major
// ---------------------------------------------------------------------------
__global__ void sdp_finalize_kernel(const float* __restrict__ accum,
                                    float* __restrict__ out) {
  int b = blockIdx.x * blockDim.x + threadIdx.x;
  if (b < NBATCH) {
    out[2 * b + 0] = accum[b];
    out[2 * b + 1] = accum[NBATCH + b];
  }
}

// ---------------------------------------------------------------------------
extern "C" void kernel_launch(void* const* d_in, const int* in_sizes, int n_in,
                              void* d_out, int out_size, void* d_ws, size_t ws_size,
                              hipStream_t stream) {
  const float* features = (const float*)d_in[0];
  const int*   bids     = (const int*)d_in[1];
  const float* W1       = (const float*)d_in[2];
  const float* b1       = (const float*)d_in[3];
  const float* W2       = (const float*)d_in[4];
  const float* b2       = (const float*)d_in[5];
  const float* W3       = (const float*)d_in[6];
  const float* b3       = (const float*)d_in[7];
  const int n = in_sizes[0] / 2;  // features is (N,2)

  float* accum = (float*)d_ws;                              // 2*NBATCH floats
  unsigned int* counts = (unsigned int*)d_ws + 2 * NBATCH;  // NBATCH u32

  hipMemsetAsync(d_ws, 0, 3 * NBATCH * sizeof(float), stream);

  sdp_count_kernel<<<(n + 255) / 256, 256, 0, stream>>>(bids, counts, n);

  sdp_main_kernel<<<n / PTS_PER_BLOCK, 256, 0, stream>>>(
      features, bids, W1, b1, W2, b2, W3, b3, counts, accum);

  sdp_finalize_kernel<<<(NBATCH + 255) / 256, 256, 0, stream>>>(accum, (float*)d_out);
}